// GCN_gen_29892972380410
// MI455X (gfx1250) — compile-verified
//
#include <hip/hip_runtime.h>
#include <hip/hip_bf16.h>

typedef float v2f __attribute__((ext_vector_type(2)));
typedef float v8f __attribute__((ext_vector_type(8)));

#define D 128

// ---------------- degree / normalization ----------------

__global__ void gcn_init_deg(float* __restrict__ deg, int n) {
    int i = blockIdx.x * blockDim.x + threadIdx.x;
    if (i < n) deg[i] = 1.0f;   // self-loop weight
}

__global__ void gcn_deg_edges(const int* __restrict__ dst, float* __restrict__ deg, int n_edges) {
    int stride = gridDim.x * blockDim.x;
    for (int e = blockIdx.x * blockDim.x + threadIdx.x; e < n_edges; e += stride)
        atomicAdd(&deg[dst[e]], 1.0f);
}

__global__ void gcn_dinv(float* __restrict__ deg, int n) {
    int i = blockIdx.x * blockDim.x + threadIdx.x;
    if (i < n) {
        float d = deg[i];
        deg[i] = (d > 0.0f) ? __frsqrt_rn(d) : 0.0f;   // in-place deg -> dinv
    }
}

// ---------------- GEMM: H = (RELU ? relu(X) : X) @ W  via f32 WMMA ----------------
// One wave computes one 16x16 tile of H. Block = 256 threads = 8 waves covering
// the 8 N-tiles of D=128; blockIdx.x selects the M-tile. RELU is a template
// parameter so the fused activation costs at most one v_max per element and
// layer 1 pays nothing.

template <bool RELU>
__global__ void gcn_gemm_wmma(const float* __restrict__ X, const float* __restrict__ W,
                              float* __restrict__ H, int n_rows) {
    const int wave = threadIdx.x >> 5;
    const int lane = threadIdx.x & 31;
    const int mbase = blockIdx.x * 16;
    const int nbase = wave * 16;
    const int half = lane >> 4;   // 0: lanes 0-15, 1: lanes 16-31
    const int l16  = lane & 15;

    const bool full_tile = (mbase + 16 <= n_rows);
    int arow = mbase + l16;
    if (arow >= n_rows) arow = n_rows - 1;     // clamp: feeds only unstored C rows
    const float* __restrict__ ap = X + (size_t)arow * D;
    const int bcol = nbase + l16;

    v8f acc = {};
    #pragma unroll 8
    for (int k = 0; k < D; k += 4) {
        const int kk = k + half * 2;
        // A fragment: 16x4, row = mbase + l16, K = kk, kk+1 (contiguous -> b64 load)
        v2f a;
        a.x = ap[kk];
        a.y = ap[kk + 1];
        if (RELU) { a.x = fmaxf(a.x, 0.0f); a.y = fmaxf(a.y, 0.0f); }
        // B fragment: 4x16, col = nbase + l16, K = kk, kk+1 (row-major W[k][n])
        v2f b;
        b.x = W[(size_t)kk * D + bcol];
        b.y = W[(size_t)(kk + 1) * D + bcol];
        acc = __builtin_amdgcn_wmma_f32_16x16x4_f32(
            /*neg_a=*/false, a, /*neg_b=*/false, b,
            /*c_mod=*/(short)0, acc, /*reuse_a=*/false, /*reuse_b=*/false);
    }

    // C/D layout: VGPR v -> M = v (lanes 0-15) or M = v+8 (lanes 16-31), N = l16
    const int crow0 = mbase + half * 8;
    float* __restrict__ cp = H + (size_t)crow0 * D + nbase + l16;
    if (full_tile) {
        #pragma unroll
        for (int v = 0; v < 8; ++v) cp[(size_t)v * D] = acc[v];
    } else {
        #pragma unroll
        for (int v = 0; v < 8; ++v)
            if (crow0 + v < n_rows) cp[(size_t)v * D] = acc[v];
    }
}

// ---------------- aggregation: self-loop + bias init, then edge scatter ----------------

__global__ void gcn_agg_init(const float* __restrict__ H, const float* __restrict__ dinv,
                             const float* __restrict__ bias, float* __restrict__ Y, int n_rows) {
    int tid = blockIdx.x * blockDim.x + threadIdx.x;
    int node = tid >> 5;            // 32 float4 chunks per node (D=128)
    int c = (tid & 31) * 4;
    if (node >= n_rows) return;
    float s = dinv[node];
    s = s * s;                      // self-loop norm = dinv[i]^2
    float4 h = *(const float4*)(H + (size_t)node * D + c);
    float4 b = *(const float4*)(bias + c);
    float4 r = make_float4(fmaf(h.x, s, b.x), fmaf(h.y, s, b.y),
                           fmaf(h.z, s, b.z), fmaf(h.w, s, b.w));
    *(float4*)(Y + (size_t)node * D + c) = r;
}

__global__ void gcn_agg_edges(const float* __restrict__ H, const float* __restrict__ dinv,
                              const int* __restrict__ src, const int* __restrict__ dst,
                              float* __restrict__ Y, int n_edges) {
    const int lane = threadIdx.x & 31;
    const int wid = (blockIdx.x * blockDim.x + threadIdx.x) >> 5;
    const int nwaves = (gridDim.x * blockDim.x) >> 5;

    for (int e = wid; e < n_edges; e += nwaves) {
        const int s = src[e];
        const int d = dst[e];
        // Prefetch next iteration's source row (L2-resident gather)
        const int e2 = e + nwaves;
        if (e2 < n_edges) {
            const int s2 = src[e2];
            __builtin_prefetch(H + (size_t)s2 * D + lane * 4, 0, 1);
        }
        const float norm = dinv[s] * dinv[d];
        float4 h = *(const float4*)(H + (size_t)s * D + lane * 4);
        float* yp = Y + (size_t)d * D + lane * 4;
        atomicAdd(yp + 0, h.x * norm);
        atomicAdd(yp + 1, h.y * norm);
        atomicAdd(yp + 2, h.z * norm);
        atomicAdd(yp + 3, h.w * norm);
    }
}

// ---------------- driver ----------------

extern "C" void kernel_launch(void* const* d_in, const int* in_sizes, int n_in,
                              void* d_out, int out_size, void* d_ws, size_t ws_size,
                              hipStream_t stream) {
    const float* x   = (const float*)d_in[0];
    const int*   ei  = (const int*)d_in[1];
    const float* W1  = (const float*)d_in[2];
    const float* b1  = (const float*)d_in[3];
    const float* W2  = (const float*)d_in[4];
    const float* b2  = (const float*)d_in[5];
    const float* W3  = (const float*)d_in[6];
    const float* b3  = (const float*)d_in[7];

    const int N = in_sizes[0] / D;
    const int E = in_sizes[1] / 2;
    const int* src = ei;
    const int* dst = ei + E;

    // workspace layout: dinv[N] | H[N*D] | Y[N*D]
    float* dinv = (float*)d_ws;
    size_t off = ((size_t)N * sizeof(float) + 255) & ~(size_t)255;
    float* H = (float*)((char*)d_ws + off);
    float* Y = H + (size_t)N * D;
    float* out = (float*)d_out;

    const int T = 256;
    const int nodeBlocks  = (N + T - 1) / T;
    const int mTiles      = (N + 15) / 16;
    const int initBlocks  = (int)(((size_t)N * 32 + T - 1) / T);
    const int edgeBlocks  = 2048;   // grid-stride, 16384 waves over E edges

    // ---- degrees (once, shared across layers) ----
    gcn_init_deg<<<nodeBlocks, T, 0, stream>>>(dinv, N);
    gcn_deg_edges<<<1024, T, 0, stream>>>(dst, dinv, E);
    gcn_dinv<<<nodeBlocks, T, 0, stream>>>(dinv, N);

    // ---- layer 1 ----
    gcn_gemm_wmma<false><<<mTiles, T, 0, stream>>>(x, W1, H, N);
    gcn_agg_init<<<initBlocks, T, 0, stream>>>(H, dinv, b1, Y, N);
    gcn_agg_edges<<<edgeBlocks, T, 0, stream>>>(H, dinv, src, dst, Y, E);

    // ---- layer 2 (ReLU fused into GEMM input read) ----
    gcn_gemm_wmma<true><<<mTiles, T, 0, stream>>>(Y, W2, H, N);
    gcn_agg_init<<<initBlocks, T, 0, stream>>>(H, dinv, b2, Y, N);
    gcn_agg_edges<<<edgeBlocks, T, 0, stream>>>(H, dinv, src, dst, Y, E);

    // ---- layer 3 (no output ReLU) ----
    gcn_gemm_wmma<true><<<mTiles, T, 0, stream>>>(Y, W3, H, N);
    gcn_agg_init<<<initBlocks, T, 0, stream>>>(H, dinv, b3, out, N);
    gcn_agg_edges<<<edgeBlocks, T, 0, stream>>>(H, dinv, src, dst, out, E);
}